// DualPrompt_15135464751642
// MI455X (gfx1250) — compile-verified
//
#include <hip/hip_runtime.h>
#include <hip/hip_bf16.h>

typedef float v2f __attribute__((ext_vector_type(2)));
typedef float v8f __attribute__((ext_vector_type(8)));
typedef float nf4 __attribute__((ext_vector_type(4)));

#define DIM   768
#define POOL  100
#define NPAD  112          // POOL padded to multiple of 16
#define NT    7            // NPAD / 16
#define BATCH 1024
#define PLEN  8
#define EK_SZ (BATCH * 4 * DIM)          // 3,145,728 floats (Ek); Ev same
#define XB_SZ (BATCH * 197 * DIM)        // 154,927,104 floats

// ---------------------------------------------------------------------------
// 1) prep: W1[k,d] = softmax(A[k,:])[d] * l2norm(K[k,:])[d] ; W2[k,d] = sm^2
//    one wave (32 lanes) per pool row; rows >= POOL are zero padding.
// ---------------------------------------------------------------------------
__global__ __launch_bounds__(32) void dp_prep(const float* __restrict__ K,
                                              const float* __restrict__ A,
                                              float* __restrict__ W1,
                                              float* __restrict__ W2) {
    const int k    = blockIdx.x;
    const int lane = threadIdx.x;
    if (k >= POOL) {
        for (int d = lane; d < DIM; d += 32) {
            W1[k * DIM + d] = 0.0f;
            W2[k * DIM + d] = 0.0f;
        }
        return;
    }
    float av[24], kv[24];
    float mx = -3.0e38f, nrm2 = 0.0f;
    #pragma unroll
    for (int i = 0; i < 24; ++i) {
        const int d = lane + 32 * i;
        av[i] = A[k * DIM + d];
        kv[i] = K[k * DIM + d];
        mx    = fmaxf(mx, av[i]);
        nrm2 += kv[i] * kv[i];
    }
    #pragma unroll
    for (int off = 16; off >= 1; off >>= 1) {
        mx    = fmaxf(mx, __shfl_xor(mx, off, 32));
        nrm2 += __shfl_xor(nrm2, off, 32);
    }
    float Z = 0.0f;
    #pragma unroll
    for (int i = 0; i < 24; ++i) { av[i] = __expf(av[i] - mx); Z += av[i]; }
    #pragma unroll
    for (int off = 16; off >= 1; off >>= 1) Z += __shfl_xor(Z, off, 32);
    const float invZ = 1.0f / Z;
    const float invN = 1.0f / fmaxf(sqrtf(nrm2), 1e-12f);
    #pragma unroll
    for (int i = 0; i < 24; ++i) {
        const int d = lane + 32 * i;
        const float s = av[i] * invZ;
        W1[k * DIM + d] = s * (kv[i] * invN);
        W2[k * DIM + d] = s * s;
    }
}

// ---------------------------------------------------------------------------
// 2) GEMM via V_WMMA_F32_16X16X4_F32: one wave per 16-row batch tile.
//    num  = Xq  x W1^T   (16 x NPAD)
//    den2 = Xq^2 x W2^T  (16 x NPAD)
//    aq[b,n] = num / max(sqrt(den2), eps); padded cols -> -3e38
// ---------------------------------------------------------------------------
__global__ __launch_bounds__(32) void dp_gemm_aq(const float* __restrict__ xq,
                                                 const float* __restrict__ W1,
                                                 const float* __restrict__ W2,
                                                 float* __restrict__ aq) {
    __shared__ float xs[16][DIM];               // 48 KB staged A-tile
    const int lane = threadIdx.x;
    const int b0   = blockIdx.x * 16;

    {   // coalesced float4 staging of the 16x768 Xq tile
        const nf4* src = (const nf4*)(xq + (size_t)b0 * DIM);
        nf4*       dst = (nf4*)(&xs[0][0]);
        for (int i = lane; i < 16 * DIM / 4; i += 32) dst[i] = src[i];
    }
    __syncthreads();

    const int half = lane >> 4;                 // 0: K pair {0,1}; 1: {2,3}
    const int m    = lane & 15;                 // A row / B column within tile

    v8f accN[NT], accD[NT];
    #pragma unroll
    for (int t = 0; t < NT; ++t) {
        accN[t] = (v8f){0,0,0,0,0,0,0,0};
        accD[t] = (v8f){0,0,0,0,0,0,0,0};
    }

    for (int kk = 0; kk < DIM; kk += 4) {
        const int ko = kk + 2 * half;
        // A-frag: lane m holds Xq[b0+m][ko], Xq[b0+m][ko+1]
        const v2f a  = *(const v2f*)(&xs[m][ko]);
        const v2f a2 = a * a;
        #pragma unroll
        for (int t = 0; t < NT; ++t) {
            const int n = t * 16 + m;
            // B-frag: lane m holds W[n][ko], W[n][ko+1]  (B = W^T)
            const v2f b1 = *(const v2f*)(W1 + (size_t)n * DIM + ko);
            const v2f b2 = *(const v2f*)(W2 + (size_t)n * DIM + ko);
            accN[t] = __builtin_amdgcn_wmma_f32_16x16x4_f32(
                false, a,  false, b1, (short)0, accN[t], false, false);
            accD[t] = __builtin_amdgcn_wmma_f32_16x16x4_f32(
                false, a2, false, b2, (short)0, accD[t], false, false);
        }
    }

    // C/D layout: lane (half,m), VGPR r -> row M = r + 8*half, col N = t*16+m
    #pragma unroll
    for (int t = 0; t < NT; ++t) {
        const int col = t * 16 + m;
        #pragma unroll
        for (int r = 0; r < 8; ++r) {
            const int row = r + 8 * half;
            const float num = accN[t][r];
            const float den = sqrtf(fmaxf(accD[t][r], 0.0f));
            float v = num / fmaxf(den, 1e-12f);
            if (col >= POOL) v = -3.0e38f;
            aq[(size_t)(b0 + row) * NPAD + col] = v;
        }
    }
}

// ---------------------------------------------------------------------------
// 3) top-3 per batch row: stable (earlier index wins ties) like jax top_k.
// ---------------------------------------------------------------------------
__global__ void dp_top3(const float* __restrict__ aq, int* __restrict__ idx) {
    const int b = blockIdx.x * blockDim.x + threadIdx.x;
    if (b >= BATCH) return;
    const float* row = aq + (size_t)b * NPAD;
    float v0 = -3.4e38f, v1 = -3.4e38f, v2 = -3.4e38f;
    int   i0 = 0, i1 = 0, i2 = 0;
    for (int k = 0; k < POOL; ++k) {
        const float v = row[k];
        if (v > v0)      { v2 = v1; i2 = i1; v1 = v0; i1 = i0; v0 = v; i0 = k; }
        else if (v > v1) { v2 = v1; i2 = i1; v1 = v;  i1 = k; }
        else if (v > v2) { v2 = v;  i2 = k; }
    }
    idx[b * 3 + 0] = i0;
    idx[b * 3 + 1] = i1;
    idx[b * 3 + 2] = i2;
}

// ---------------------------------------------------------------------------
// 4) gather+sum: Ek[b,j,d] (j<4) / Ev[b,j-4,d] (j>=4) = sum_t p[j, idx[b,t], d]
//    one block per (b, j) pair; p is 2.4 MB -> L2 resident.
// ---------------------------------------------------------------------------
__global__ void dp_gather(const float* __restrict__ p,
                          const int* __restrict__ idx,
                          float* __restrict__ out) {
    const int blk = blockIdx.x;           // 0 .. BATCH*PLEN-1
    const int b   = blk >> 3;
    const int j   = blk & 7;
    const int t0 = idx[b * 3 + 0];
    const int t1 = idx[b * 3 + 1];
    const int t2 = idx[b * 3 + 2];
    const float* p0 = p + ((size_t)j * POOL + t0) * DIM;
    const float* p1 = p + ((size_t)j * POOL + t1) * DIM;
    const float* p2 = p + ((size_t)j * POOL + t2) * DIM;
    float* dst = (j < 4) ? out + ((size_t)b * 4 + j) * DIM
                         : out + (size_t)EK_SZ + ((size_t)b * 4 + (j - 4)) * DIM;
    for (int d = threadIdx.x; d < DIM; d += blockDim.x)
        dst[d] = p0[d] + p1[d] + p2[d];
}

// ---------------------------------------------------------------------------
// 5) x_block pass-through: 620 MB, non-temporal float4 streaming copy
//    (working set >> 192 MB L2; NT hints keep L2 clean for p/W1/W2).
// ---------------------------------------------------------------------------
__global__ void dp_copy(const nf4* __restrict__ src, nf4* __restrict__ dst,
                        long n4) {
    long i = (long)blockIdx.x * blockDim.x + threadIdx.x;
    const long stride = (long)gridDim.x * blockDim.x;
    for (; i < n4; i += stride) {
        nf4 v = __builtin_nontemporal_load(&src[i]);
        __builtin_nontemporal_store(v, &dst[i]);
    }
}

extern "C" void kernel_launch(void* const* d_in, const int* in_sizes, int n_in,
                              void* d_out, int out_size, void* d_ws, size_t ws_size,
                              hipStream_t stream) {
    const float* xq = (const float*)d_in[0];   // (1024, 768)
    const float* xb = (const float*)d_in[1];   // (1024, 197, 768)
    const float* K  = (const float*)d_in[2];   // (100, 768)
    const float* A  = (const float*)d_in[3];   // (100, 768)
    const float* p  = (const float*)d_in[4];   // (8, 100, 768)
    // d_in[5] = l (unused by the math)

    float* out = (float*)d_out;
    float* ws  = (float*)d_ws;
    float* W1  = ws;                               // 112*768
    float* W2  = W1 + NPAD * DIM;                  // 112*768
    float* aq  = W2 + NPAD * DIM;                  // 1024*112
    int*   idx = (int*)(aq + (size_t)BATCH * NPAD);// 1024*3

    dp_prep   <<<NPAD, 32, 0, stream>>>(K, A, W1, W2);
    dp_gemm_aq<<<BATCH / 16, 32, 0, stream>>>(xq, W1, W2, aq);
    dp_top3   <<<(BATCH + 255) / 256, 256, 0, stream>>>(aq, idx);
    dp_gather <<<BATCH * PLEN, 256, 0, stream>>>(p, idx, out);
    dp_copy   <<<4096, 256, 0, stream>>>((const nf4*)xb,
                                         (nf4*)(out + 2 * (size_t)EK_SZ),
                                         (long)XB_SZ / 4);
}